// GCNLayer_81157702025495
// MI455X (gfx1250) — compile-verified
//
#include <hip/hip_runtime.h>

#define D 128  // D_IN == D_OUT == 128

typedef float v2f __attribute__((ext_vector_type(2)));
typedef float v8f __attribute__((ext_vector_type(8)));

// ---------------------------------------------------------------------------
// Kernel 1: zero the aggregation scratch (float4-vectorized)
// ---------------------------------------------------------------------------
__global__ void gcn_zero_kernel(float4* __restrict__ p, int n4) {
    int i = blockIdx.x * blockDim.x + threadIdx.x;
    if (i < n4) {
        float4 z; z.x = 0.f; z.y = 0.f; z.z = 0.f; z.w = 0.f;
        p[i] = z;
    }
}

// ---------------------------------------------------------------------------
// Kernel 2: edge scatter-add.  One wave32 per edge: each lane moves 4
// consecutive floats of the 128-float row (float4 gather, 4x f32 atomics).
// x and agg are ~51MB each -> L2-resident on MI455X (192MB L2), so the
// gather + atomic traffic never goes to HBM after first touch.
// ---------------------------------------------------------------------------
__global__ void __launch_bounds__(256)
gcn_scatter_kernel(const float* __restrict__ x,
                   const long long* __restrict__ ei,  // int64 edge_index (2,E)
                   float* __restrict__ agg, int E) {
    int e    = blockIdx.x * 8 + (threadIdx.x >> 5);   // 8 edges per 256-thr block
    int lane = threadIdx.x & 31;
    if (e >= E) return;
    long long src = ei[e];
    long long dst = ei[(long long)E + e];
    const float4 v = *(const float4*)(x + src * D + lane * 4);
    float* a = agg + dst * D + lane * 4;
    atomicAdd(a + 0, v.x);   // global_atomic_add_f32, no-return (STOREcnt)
    atomicAdd(a + 1, v.y);
    atomicAdd(a + 2, v.z);
    atomicAdd(a + 3, v.w);
}

// ---------------------------------------------------------------------------
// Kernel 3: out = agg @ W + b using V_WMMA_F32_16X16X4_F32 (f32 matrix pipe).
// Block = 256 threads = 8 waves. Block owns 16 output rows; wave w owns
// output cols [16w, 16w+16). A-slab (16x128, 8KB) staged in LDS and shared
// by all 8 waves; each wave preloads its B column slab (128x16 of W) into
// 32 v2f fragments, then chains 32 WMMAs over K.
//
// ISA 7.12.2 layouts (32-bit, wave32):
//   A 16x4 : M = lane%16,  K = 2*(lane/16) + j      (j = vgpr 0/1)
//   B 4x16 : N = lane%16,  K = 2*(lane/16) + j
//   C 16x16: N = lane%16,  M = vgpr + 8*(lane/16)
// ---------------------------------------------------------------------------
__global__ void __launch_bounds__(256)
gcn_gemm_kernel(const float* __restrict__ agg, const float* __restrict__ Wm,
                const float* __restrict__ bias, float* __restrict__ out,
                int nrows) {
    __shared__ float As[16 * D];   // 8 KB

    const int rbase   = blockIdx.x * 16;
    const int wave    = threadIdx.x >> 5;
    const int lane    = threadIdx.x & 31;
    const int hl      = lane >> 4;      // lane half (0/1)
    const int l16     = lane & 15;
    const int colbase = wave * 16;

    // ---- stage A block (16 rows x 128) into LDS: 512 float4, 2 per thread
    {
        const float4* s4 = (const float4*)(agg + (size_t)rbase * D);
        float4* d4 = (float4*)As;
        d4[threadIdx.x]       = s4[threadIdx.x];
        d4[threadIdx.x + 256] = s4[threadIdx.x + 256];
    }
    __syncthreads();

    // ---- preload B fragments: wave's 128x16 slab of W -> 32 x v2f (64 VGPRs)
    v2f bfrag[32];
#pragma unroll
    for (int s = 0; s < 32; ++s) {
        const int k = s * 4 + hl * 2;
        bfrag[s][0] = Wm[(size_t)(k + 0) * D + colbase + l16];
        bfrag[s][1] = Wm[(size_t)(k + 1) * D + colbase + l16];
    }

    // ---- K loop: 32 chained f32 WMMAs (K = 128 in steps of 4)
    v8f c = {0.f, 0.f, 0.f, 0.f, 0.f, 0.f, 0.f, 0.f};
#pragma unroll
    for (int s = 0; s < 32; ++s) {
        const int k = s * 4 + hl * 2;
        v2f a = *(const v2f*)(As + l16 * D + k);   // ds_load_b64
        // 8 args: (neg_a, A, neg_b, B, c_mod, C, reuse_a, reuse_b)
        c = __builtin_amdgcn_wmma_f32_16x16x4_f32(
                false, a, false, bfrag[s], (short)0, c, false, false);
    }

    // ---- epilogue: add bias, store 16x16 tile
    const float bb = bias[colbase + l16];
#pragma unroll
    for (int v = 0; v < 8; ++v) {
        const int row = rbase + v + hl * 8;
        if (row < nrows)
            out[(size_t)row * D + colbase + l16] = c[v] + bb;
    }
}

// ---------------------------------------------------------------------------
extern "C" void kernel_launch(void* const* d_in, const int* in_sizes, int n_in,
                              void* d_out, int out_size, void* d_ws, size_t ws_size,
                              hipStream_t stream) {
    const float*     x  = (const float*)d_in[0];      // (N, 128) f32
    const long long* ei = (const long long*)d_in[1];  // (2, E) int64
    const float*     Wm = (const float*)d_in[2];      // (128, 128) f32
    const float*     b  = (const float*)d_in[3];      // (128,) f32
    float* out = (float*)d_out;
    float* agg = (float*)d_ws;                        // N*128 f32 scratch (51.2 MB)

    const int ND = in_sizes[0];        // N * 128
    const int N  = ND / D;
    const int E  = in_sizes[1] / 2;

    // 1) agg = 0
    const int n4 = ND / 4;
    gcn_zero_kernel<<<(n4 + 255) / 256, 256, 0, stream>>>((float4*)agg, n4);

    // 2) agg[dst] += x[src]   (one wave per edge)
    gcn_scatter_kernel<<<(E + 7) / 8, 256, 0, stream>>>(x, ei, agg, E);

    // 3) out = agg @ W + b    (f32 WMMA)
    gcn_gemm_kernel<<<(N + 15) / 16, 256, 0, stream>>>(agg, Wm, b, out, N);
}